// DWT3DInverse_66400194396247
// MI455X (gfx1250) — compile-verified
//
#include <hip/hip_runtime.h>

typedef float v2f __attribute__((ext_vector_type(2)));
typedef float v8f __attribute__((ext_vector_type(8)));

#define U_STR 25  // padded u-vector stride (floats); odd -> LDS-bank friendly

// Banded synthesis coefficient: W[m, s'] for output row m (block-local) and
// u-window position s' = u_loc - 4 in [-4,20).  q parity picks g0 (lo) / g1 (hi).
__device__ __forceinline__ float wcoef(const float* __restrict__ g0,
                                       const float* __restrict__ g1,
                                       int m, int s) {
  if (s & 1) { int k = m + 4 - s; return (k >= 0 && k < 8) ? g1[k] : 0.0f; }
  else       { int k = m + 3 - s; return (k >= 0 && k < 8) ? g0[k] : 0.0f; }
}

__global__ __launch_bounds__(256)
void idwt3_level_kernel(const float* __restrict__ ll, long long ll_bstride,
                        const float* __restrict__ hb, long long h_bstride,
                        float* __restrict__ out,
                        const float* __restrict__ g0,
                        const float* __restrict__ g1,
                        int n)
{
  const int nb = (2 * n) / 16;            // output tile-blocks per axis
  int t = blockIdx.x;
  const int bw = t % nb; t /= nb;
  const int bh = t % nb; t /= nb;
  const int bd = t;
  const int batch = blockIdx.y;

  const int tid  = threadIdx.x;
  const int lane = tid & 31;
  const int wave = tid >> 5;              // 0..7
  const int mrow = lane & 15;             // M / N index within fragment
  const int kp   = lane >> 4;             // K half-select (A/B layout)

  // LDS: uB holds stage-A output (H-axis u vectors, 2 pairs x 12 d x 16 jw).
  // uAC is aliased: stage-A input pair (144 cols) then stage-B output (256 cols).
  __shared__ float uB[2 * 12 * 16 * U_STR];   // 9600 floats
  __shared__ float uAC[16 * 16 * U_STR];      // 6400 floats
  float* uA1 = uAC;                           // 144*U_STR = 3600 <= 6400
  float* uC  = uAC;

  // ---- Build the six 16x4 A-matrix fragments (shared by all axes/levels) ----
  // f32 16x4 A layout: lanes 0-15 -> M=lane, {v0,v1}={K0,K1}; lanes 16-31 -> {K2,K3}.
  float ax[6], ay[6];
#pragma unroll
  for (int tt = 0; tt < 6; ++tt) {
    int s0 = 4 * tt - 4 + 2 * kp;         // even
    ax[tt] = wcoef(g0, g1, mrow, s0);
    ay[tt] = wcoef(g0, g1, mrow, s0 + 1); // odd
  }

  const long long nn = (long long)n * n;
  const long long n3 = nn * n;
  const int iw0 = 8 * bw - 2, ih0 = 8 * bh - 2, id0 = 8 * bd - 2;

  // ================= Stage A: synthesize along W, pairs (c, c+4) =============
  for (int c = 0; c < 4; ++c) {
    __syncthreads();                       // uA1 / uAC reuse guard
    // Fill interleaved u for this pair: uA1[(d*12+h)*U + 2*w + p], p=0 lo / 1 hi
    for (int idx = tid; idx < 12 * 12 * 24; idx += 256) {
      int q = idx % 24;
      int h = (idx / 24) % 12;
      int d = idx / (24 * 12);
      int w = q >> 1, p = q & 1;
      int s = c + 4 * p;                   // global subband (bit2 = W selector)
      int iw = (iw0 + w + n) % n;
      int ih = (ih0 + h + n) % n;
      int id = (id0 + d + n) % n;
      long long off = (long long)id * nn + (long long)ih * n + iw;
      float v = (s == 0)
                  ? ll[(long long)batch * ll_bstride + off]
                  : hb[(long long)batch * h_bstride + (long long)(s - 1) * n3 + off];
      uA1[(d * 12 + h) * U_STR + q] = v;
    }
    __syncthreads();

    // 144 columns = 9 groups of 16; C rows = jw, cols = (d,h)
    for (int grp = wave; grp < 9; grp += 8) {
      v8f acc = {0.f, 0.f, 0.f, 0.f, 0.f, 0.f, 0.f, 0.f};
      int col = grp * 16 + mrow;
      int dl = col / 12, hl = col % 12;
      const float* ucol = &uA1[(dl * 12 + hl) * U_STR];
#pragma unroll
      for (int tt = 0; tt < 6; ++tt) {
        v2f av; av.x = ax[tt]; av.y = ay[tt];
        v2f bv; bv.x = ucol[4 * tt + 2 * kp]; bv.y = ucol[4 * tt + 2 * kp + 1];
        acc = __builtin_amdgcn_wmma_f32_16x16x4_f32(
            false, av, false, bv, (short)0, acc, false, false);
      }
      int par = (c >> 1) & 1;              // bit1 = H selector
      int g   = c & 1;                     // bit0 = D selector
#pragma unroll
      for (int v = 0; v < 8; ++v) {
        int jw = v + 8 * kp;               // C layout: M = v + 8*(lane>>4)
        uB[((g * 12 + dl) * 16 + jw) * U_STR + 2 * hl + par] = acc[v];
      }
    }
  }
  __syncthreads();

  // ================= Stage B: synthesize along H ============================
  // 2 pairs x 192 columns (d,jw) = 24 tasks; C rows = jh
  for (int task = wave; task < 24; task += 8) {
    int g = task / 12;
    int grp = task % 12;
    v8f acc = {0.f, 0.f, 0.f, 0.f, 0.f, 0.f, 0.f, 0.f};
    int col = grp * 16 + mrow;
    int dl = col / 16, jw = col % 16;
    const float* ucol = &uB[((g * 12 + dl) * 16 + jw) * U_STR];
#pragma unroll
    for (int tt = 0; tt < 6; ++tt) {
      v2f av; av.x = ax[tt]; av.y = ay[tt];
      v2f bv; bv.x = ucol[4 * tt + 2 * kp]; bv.y = ucol[4 * tt + 2 * kp + 1];
      acc = __builtin_amdgcn_wmma_f32_16x16x4_f32(
          false, av, false, bv, (short)0, acc, false, false);
    }
#pragma unroll
    for (int v = 0; v < 8; ++v) {
      int jh = v + 8 * kp;
      uC[(jh * 16 + jw) * U_STR + 2 * dl + g] = acc[v];
    }
  }
  __syncthreads();

  // ================= Stage C: synthesize along D, write global ==============
  const int N2 = 2 * n;
  const long long N2sq = (long long)N2 * N2;
  const long long obat = (long long)batch * N2 * N2sq;
  for (int grp = wave; grp < 16; grp += 8) {
    v8f acc = {0.f, 0.f, 0.f, 0.f, 0.f, 0.f, 0.f, 0.f};
    int col = grp * 16 + mrow;
    int jh = col / 16, jw = col % 16;
    const float* ucol = &uC[(jh * 16 + jw) * U_STR];
#pragma unroll
    for (int tt = 0; tt < 6; ++tt) {
      v2f av; av.x = ax[tt]; av.y = ay[tt];
      v2f bv; bv.x = ucol[4 * tt + 2 * kp]; bv.y = ucol[4 * tt + 2 * kp + 1];
      acc = __builtin_amdgcn_wmma_f32_16x16x4_f32(
          false, av, false, bv, (short)0, acc, false, false);
    }
    long long base = obat + (long long)(16 * bh + jh) * N2 + (16 * bw + jw);
#pragma unroll
    for (int v = 0; v < 8; ++v) {
      int jd = 16 * bd + v + 8 * kp;
      out[base + (long long)jd * N2sq] = acc[v];
    }
  }
}

extern "C" void kernel_launch(void* const* d_in, const int* in_sizes, int n_in,
                              void* d_out, int out_size, void* d_ws, size_t ws_size,
                              hipStream_t stream) {
  (void)in_sizes; (void)n_in; (void)out_size; (void)ws_size;
  const float* yl  = (const float*)d_in[0];   // (4,1,64,64,64)
  const float* yh1 = (const float*)d_in[1];   // (4,7,128,128,128)
  const float* yh2 = (const float*)d_in[2];   // (4,7,64,64,64)
  const float* g0  = (const float*)d_in[3];
  const float* g1  = (const float*)d_in[4];
  float* LL = (float*)d_ws;                   // 4 * 128^3 floats = 32 MiB

  {   // Level 1: n = 64 -> LL (4,128^3)
    const int n = 64, nb = (2 * n) / 16;      // 8
    dim3 grid(nb * nb * nb, 4);
    idwt3_level_kernel<<<grid, 256, 0, stream>>>(
        yl, (long long)n * n * n, yh2, 7LL * n * n * n, LL, g0, g1, n);
  }
  {   // Level 2: n = 128 -> d_out (4,256^3)
    const int n = 128, nb = (2 * n) / 16;     // 16
    dim3 grid(nb * nb * nb, 4);
    idwt3_level_kernel<<<grid, 256, 0, stream>>>(
        LL, (long long)n * n * n, yh1, 7LL * n * n * n,
        (float*)d_out, g0, g1, n);
  }
}